// layer_last_26113401160398
// MI455X (gfx1250) — compile-verified
//
#include <hip/hip_runtime.h>

typedef __attribute__((ext_vector_type(2))) float v2f;
typedef __attribute__((ext_vector_type(8))) float v8f;

#define BS     32
#define NN     512
#define JJ     4
#define NFEAT  64
#define NOUT   128
#define CDIM   (JJ * NFEAT)      // 256
#define NSPLIT 8
#define NCHUNK (NN / NSPLIT)     // 64

// ---------------------------------------------------------------------------
// Kernel A: S-partials. P[b][s][m][j] = sum_{n in chunk s} W[b][n][m][j]
// W innermost dim is j (J=4) -> one float4 (global_load_b128) per (b,n,m).
// 256 blocks x 512 threads = 4096 waves streaming 134 MB of W once.
// ---------------------------------------------------------------------------
__global__ __launch_bounds__(512) void colsum_kernel(const float* __restrict__ W,
                                                     float* __restrict__ P) {
    const int b = blockIdx.x / NSPLIT;
    const int s = blockIdx.x % NSPLIT;
    const int m = threadIdx.x;                    // 0..511
    const float4* Wp = (const float4*)W;          // float4 over j
    float4 acc = make_float4(0.f, 0.f, 0.f, 0.f);
    const size_t base = ((size_t)b * NN + (size_t)s * NCHUNK) * NN + m;
#pragma unroll 8
    for (int i = 0; i < NCHUNK; ++i) {
        float4 w = Wp[base + (size_t)i * NN];
        acc.x += w.x; acc.y += w.y; acc.z += w.z; acc.w += w.w;
    }
    ((float4*)P)[((size_t)b * NSPLIT + s) * NN + m] = acc;
}

// ---------------------------------------------------------------------------
// Kernel B1: reduce partials, transposing to j-major: ST[b][j][m]
// (j-major makes the GEMM1 B-fragment a single b64 load per K-pair)
// ---------------------------------------------------------------------------
__global__ __launch_bounds__(256) void reduce_kernel(const float* __restrict__ P,
                                                     float* __restrict__ ST) {
    const int idx = blockIdx.x * 256 + threadIdx.x;   // over BS*NN*JJ = 65536
    const int b  = idx / (NN * JJ);
    const int mj = idx % (NN * JJ);                   // mj = m*4 + j
    float acc = 0.f;
#pragma unroll
    for (int s = 0; s < NSPLIT; ++s)
        acc += P[((size_t)b * NSPLIT + s) * (NN * JJ) + mj];
    const int m = mj >> 2;
    const int j = mj & 3;
    ST[((size_t)b * JJ + j) * NN + m] = acc;
}

// ---------------------------------------------------------------------------
// Kernel B2: per-batch GEMM  t2[b, n*64+f] = sum_m X[b,f,m] * ST[b,n,m]
// 64x16x512 per batch via V_WMMA_F32_16X16X4_F32 (N padded 4->16).
// Padded columns n>=4 read clamped (n&3) addresses: their D columns are
// garbage but never stored (D columns are independent). No predication in
// the inner loop; two accumulators halve the dependent WMMA chain.
// ---------------------------------------------------------------------------
__global__ __launch_bounds__(128) void gemm1_wmma(const float* __restrict__ X,
                                                  const float* __restrict__ ST,
                                                  float* __restrict__ T2) {
    const int b    = blockIdx.x;
    const int wave = threadIdx.x >> 5;       // 0..3
    const int lane = threadIdx.x & 31;
    const int f0   = wave * 16;
    const int hl   = lane & 15;
    const int kb   = (lane < 16) ? 0 : 2;    // A/B K-offset per half-wave
    const int n    = hl;                     // B column (only n<4 is real)

    const float* Xb = X  + (size_t)b * NFEAT * NN + (size_t)(f0 + hl) * NN + kb;
    const float* Sn = ST + (size_t)b * JJ * NN    + (size_t)(n & 3) * NN   + kb;

    v8f acc0 = {}, acc1 = {};
    for (int m0 = 0; m0 < NN; m0 += 8) {
        v2f a0 = *(const v2f*)(Xb + m0);
        v2f b0 = *(const v2f*)(Sn + m0);
        v2f a1 = *(const v2f*)(Xb + m0 + 4);
        v2f b1 = *(const v2f*)(Sn + m0 + 4);
        acc0 = __builtin_amdgcn_wmma_f32_16x16x4_f32(
            false, a0, false, b0, (short)0, acc0, false, false);
        acc1 = __builtin_amdgcn_wmma_f32_16x16x4_f32(
            false, a1, false, b1, (short)0, acc1, false, false);
    }

    if (n < JJ) {
        const int mofs = (lane < 16) ? 0 : 8;    // C/D: M = p + mofs
#pragma unroll
        for (int p = 0; p < 8; ++p) {
            const int f = f0 + p + mofs;
            T2[(size_t)b * CDIM + n * NFEAT + f] = acc0[p] + acc1[p];
        }
    }
}

// ---------------------------------------------------------------------------
// Kernel C: y[b,o] = sum_c t2[b,c]*fc_w[o,c] + 512*fc_b[o]
// 32x128x256 GEMM, one block of 16 waves (2 M-tiles x 8 N-tiles), WMMA f32.
// ---------------------------------------------------------------------------
__global__ __launch_bounds__(512) void gemm2_wmma(const float* __restrict__ T2,
                                                  const float* __restrict__ fcw,
                                                  const float* __restrict__ fcb,
                                                  float* __restrict__ Y) {
    const int wave = threadIdx.x >> 5;   // 0..15
    const int lane = threadIdx.x & 31;
    const int mt   = wave >> 3;          // 0..1  (batch tile)
    const int nt   = wave & 7;           // 0..7  (output tile)
    const int hl   = lane & 15;
    const int kb   = (lane < 16) ? 0 : 2;
    const int brow = mt * 16 + hl;       // A row (batch index)
    const int o    = nt * 16 + hl;       // B column (output channel)

    const float* Ta = T2  + (size_t)brow * CDIM + kb;
    const float* Wb = fcw + (size_t)o    * CDIM + kb;   // fc_w[o, :]

    v8f acc0 = {}, acc1 = {};
    for (int k0 = 0; k0 < CDIM; k0 += 8) {
        v2f a0 = *(const v2f*)(Ta + k0);
        v2f b0 = *(const v2f*)(Wb + k0);
        v2f a1 = *(const v2f*)(Ta + k0 + 4);
        v2f b1 = *(const v2f*)(Wb + k0 + 4);
        acc0 = __builtin_amdgcn_wmma_f32_16x16x4_f32(
            false, a0, false, b0, (short)0, acc0, false, false);
        acc1 = __builtin_amdgcn_wmma_f32_16x16x4_f32(
            false, a1, false, b1, (short)0, acc1, false, false);
    }

    const float bias = fcb[o] * (float)NN;   // sum over n of fc_b[o]
    const int mofs = (lane < 16) ? 0 : 8;
#pragma unroll
    for (int p = 0; p < 8; ++p) {
        const int brow_out = mt * 16 + p + mofs;
        Y[(size_t)brow_out * NOUT + o] = acc0[p] + acc1[p] + bias;
    }
}

// ---------------------------------------------------------------------------
extern "C" void kernel_launch(void* const* d_in, const int* in_sizes, int n_in,
                              void* d_out, int out_size, void* d_ws, size_t ws_size,
                              hipStream_t stream) {
    const float* X   = (const float*)d_in[0];   // [32, 64, 512]
    const float* W   = (const float*)d_in[1];   // [32, 512, 512, 4]
    const float* fcw = (const float*)d_in[2];   // [128, 256]
    const float* fcb = (const float*)d_in[3];   // [128]
    float* Y = (float*)d_out;                   // [32, 128]

    float* P  = (float*)d_ws;                               // 32*8*512*4  = 2 MB
    float* ST = P  + (size_t)BS * NSPLIT * NN * JJ;         // 32*4*512    = 256 KB
    float* T2 = ST + (size_t)BS * JJ * NN;                  // 32*256      = 32 KB

    colsum_kernel<<<BS * NSPLIT, 512, 0, stream>>>(W, P);
    reduce_kernel<<<(BS * NN * JJ) / 256, 256, 0, stream>>>(P, ST);
    gemm1_wmma<<<BS, 128, 0, stream>>>(X, ST, T2);
    gemm2_wmma<<<1, 512, 0, stream>>>(T2, fcw, fcb, Y);
}